// GATv2_28930899706050
// MI455X (gfx1250) — compile-verified
//
#include <hip/hip_runtime.h>
#include <hip/hip_bf16.h>
#include <math.h>
#include <stdint.h>

// ---------------------------------------------------------------------------
// GATv2 (2 layers) for MI455X / gfx1250.
//   Dense GEMMs  -> V_WMMA_F32_16X16X4_F32; weights staged into LDS by the
//                   Tensor Data Mover (tensor_load_to_lds), double-buffered
//                   and overlapped with WMMA compute (TENSORcnt + barriers).
//   Edge phases  -> L2-resident gathers + float atomics (segment softmax)
// ---------------------------------------------------------------------------

typedef __attribute__((ext_vector_type(2))) float v2f;
typedef __attribute__((ext_vector_type(8))) float v8f;
typedef unsigned int u32x4 __attribute__((ext_vector_type(4)));
typedef int i32x4 __attribute__((ext_vector_type(4)));
typedef int i32x8 __attribute__((ext_vector_type(8)));

// Order-preserving float -> uint encoding for atomicMax-based segment max.
__device__ __forceinline__ unsigned fenc(float f) {
    unsigned u = __float_as_uint(f);
    return (u & 0x80000000u) ? ~u : (u | 0x80000000u);
}
__device__ __forceinline__ float fdec(unsigned u) {
    return (u & 0x80000000u) ? __uint_as_float(u & 0x7fffffffu)
                             : __uint_as_float(~u);
}
#define ENC_NEG_INF 0x007FFFFFu   // fenc(-inf) = ~0xFF800000

#define GEMM_WAVES 8              // 256-thread blocks

// ---------------------------------------------------------------------------
// C[M,NCOLS] = A[M,K] @ W[K,NCOLS] + bias  (row-major fp32, WMMA 16x16x4 f32)
//  - W staged into LDS row-major [KC][RS] by TDM, double-buffered per K-chunk.
//  - N=64: TDM pad (4 dwords / 64) -> RS=68, conflict-free b-frag ds reads.
//  - Wave w owns rows [(bx*8+w)*16, +16); NT accumulators cover all columns.
// ---------------------------------------------------------------------------
template <int NCOLS, int K>
__global__ void wmma_gemm_bias(const float* __restrict__ A,
                               const float* __restrict__ W,
                               const float* __restrict__ bias,
                               float* __restrict__ C,
                               int M) {
    constexpr int  NT     = (NCOLS + 15) / 16;
    constexpr int  KC     = (K < 128) ? K : 128;     // K % KC == 0 here
    constexpr int  NCHUNK = K / KC;
    constexpr bool PAD    = (NCOLS % 64 == 0);       // break 64-bank alignment
    constexpr int  RS     = NCOLS + (PAD ? 4 : 0);   // LDS row stride (floats)
    constexpr int  BUF    = KC * RS;                 // floats per buffer

    extern __shared__ float lds[];                   // [2][KC][RS]
    const int lane = threadIdx.x & 31;
    const int wave = threadIdx.x >> 5;
    const int m0   = (blockIdx.x * GEMM_WAVES + wave) * 16;
    const int mr   = lane & 15;
    const int hi   = lane >> 4;
    const bool mok = (m0 < M);                       // wave-uniform

    const uint32_t lds_base = (uint32_t)(uintptr_t)(void*)lds;   // LDS byte offset

    // --- issue one TDM chunk copy: W[kc..kc+KC) x [0..NCOLS) -> lds buf ---
    auto tdm_issue = [&](int chunk, int bufsel) {
#if __has_builtin(__builtin_amdgcn_tensor_load_to_lds)
        const uint64_t ga = (uint64_t)(uintptr_t)(W + (size_t)chunk * KC * NCOLS);
        u32x4 g0;
        g0[0] = 1u;                                   // count=1, user mode
        g0[1] = lds_base + (uint32_t)(bufsel * BUF * 4);
        g0[2] = (uint32_t)ga;                         // global_addr[31:0]
        g0[3] = ((uint32_t)(ga >> 32) & 0x01FFFFFFu)  // global_addr[56:32]
              | (2u << 30);                           // type = 2 (image)
        i32x8 g1;
        g1[0] = (int)((2u << 16)                       // data_size = 4B
              | (PAD ? (1u << 20) : 0u)                // pad_enable
              | (PAD ? (5u << 22) : 0u)                // pad_interval: 64 dw
              | (PAD ? (3u << 25) : 0u));              // pad_amount: 4 dw
        g1[1] = (int)((uint32_t)NCOLS << 16);          // tensor_dim0[15:0]
        g1[2] = (int)(((uint32_t)NCOLS >> 16) | ((uint32_t)KC << 16)); // d0 hi | d1 lo
        g1[3] = (int)(((uint32_t)KC >> 16) | ((uint32_t)NCOLS << 16)); // d1 hi | tile_dim0
        g1[4] = (int)(uint32_t)KC;                     // tile_dim1 (tile_dim2=0)
        g1[5] = (int)(uint32_t)NCOLS;                  // tensor_dim0_stride[31:0]
        g1[6] = 0;                                     // stride hi | dim1_stride lo
        g1[7] = 0;
        i32x4 z4 = {};
        i32x8 z8 = {};
        // 6-arg variant (clang-23 / therock-10.0): (g0, g1, g2, g3, g4, cpol)
        __builtin_amdgcn_tensor_load_to_lds(g0, g1, z4, z4, z8, 0);
#else
        // fallback: manual staging (same LDS layout)
        for (int i = threadIdx.x & 31; i < KC * NCOLS; i += 32) {
            const int n = i % NCOLS, k = i / NCOLS;
            lds[bufsel * BUF + k * RS + n] = W[(size_t)(chunk * KC + k) * NCOLS + n];
        }
#endif
    };

    if (wave == 0) {
        tdm_issue(0, 0);
#if __has_builtin(__builtin_amdgcn_s_wait_tensorcnt)
        __builtin_amdgcn_s_wait_tensorcnt(0);
#endif
    }
    __syncthreads();

    v8f acc[NT];
    #pragma unroll
    for (int t = 0; t < NT; ++t) acc[t] = (v8f){};

    const float* __restrict__ arp = A + (size_t)(m0 + mr) * K;

    for (int c = 0; c < NCHUNK; ++c) {
        const float* __restrict__ buf = lds + (c & 1) * BUF;
        if (wave == 0 && c + 1 < NCHUNK)
            tdm_issue(c + 1, (c + 1) & 1);             // overlap with compute

        if (mok) {
            const int kc = c * KC;
            if (c + 1 < NCHUNK)
                __builtin_prefetch(arp + kc + KC, 0, 1);  // global_prefetch_b8
            for (int k0 = 0; k0 < KC; k0 += 4) {
                const int ka = k0 + (hi << 1);
                v2f a;                                  // A frag (ISA 7.12.2)
                a.x = arp[kc + ka];
                a.y = arp[kc + ka + 1];
                #pragma unroll
                for (int t = 0; t < NT; ++t) {
                    const int col  = t * 16 + mr;
                    const int colc = (col < NCOLS) ? col : (NCOLS - 1);
                    const float bx = buf[ka * RS + colc];
                    const float by = buf[(ka + 1) * RS + colc];
                    v2f b;
                    b.x = (col < NCOLS) ? bx : 0.0f;   // const-folds unless last
                    b.y = (col < NCOLS) ? by : 0.0f;   //   tile is partial
                    acc[t] = __builtin_amdgcn_wmma_f32_16x16x4_f32(
                        false, a, false, b, (short)0, acc[t], false, false);
                }
            }
        }
        __syncthreads();                               // done reading buf[c&1]
        if (c + 1 < NCHUNK) {
            if (wave == 0) {
#if __has_builtin(__builtin_amdgcn_s_wait_tensorcnt)
                __builtin_amdgcn_s_wait_tensorcnt(0);
#endif
            }
            __syncthreads();                           // buf[(c+1)&1] ready
        }
    }

    if (mok) {
        #pragma unroll
        for (int t = 0; t < NT; ++t) {
            const int col = t * 16 + mr;
            if (col < NCOLS) {
                const float bb = bias[col];
                #pragma unroll
                for (int r = 0; r < 8; ++r) {
                    const int row = m0 + r + (hi << 3);
                    C[(size_t)row * NCOLS + col] = acc[t][r] + bb;
                }
            }
        }
    }
}

// ---------------------------------------------------------------------------
// Edge pass 1: logits[e,h] = sum_d leaky_relu(hs[src,h,d]+hd[dst,h,d])*attn[h,d]
//              + atomic segment-max over dst.       (one thread per edge,head)
// ---------------------------------------------------------------------------
template <int H, int D>
__global__ void edge_logits_kernel(const int* __restrict__ src,
                                   const int* __restrict__ dst,
                                   const float* __restrict__ hs,
                                   const float* __restrict__ hd,
                                   const float* __restrict__ attn,
                                   float* __restrict__ logits,
                                   unsigned* __restrict__ maxenc,
                                   int E) {
    const long long t = (long long)blockIdx.x * blockDim.x + threadIdx.x;
    if (t >= (long long)E * H) return;
    const int e = (int)(t / H);
    const int h = (int)(t % H);
    const int sn = src[e];
    const int dn = dst[e];
    const float* __restrict__ ps = hs + ((size_t)sn * H + h) * D;
    const float* __restrict__ pd = hd + ((size_t)dn * H + h) * D;
    const float* __restrict__ pa = attn + (size_t)h * D;
    float acc = 0.0f;
    #pragma unroll
    for (int i = 0; i < D; ++i) {
        float v = ps[i] + pd[i];
        v = (v > 0.0f) ? v : 0.2f * v;        // leaky_relu, slope 0.2
        acc = fmaf(v, pa[i], acc);
    }
    logits[t] = acc;
    atomicMax(&maxenc[(size_t)dn * H + h], fenc(acc));
}

// ---------------------------------------------------------------------------
// Edge pass 2: p = exp(logit - max[dst]);  s[dst] += p   (p overwrites logits)
// ---------------------------------------------------------------------------
template <int H>
__global__ void edge_expsum_kernel(const int* __restrict__ dst,
                                   const unsigned* __restrict__ maxenc,
                                   float* __restrict__ logits,
                                   float* __restrict__ ssum,
                                   int E) {
    const long long t = (long long)blockIdx.x * blockDim.x + threadIdx.x;
    if (t >= (long long)E * H) return;
    const int e = (int)(t / H);
    const int h = (int)(t % H);
    const int dn = dst[e];
    const float m = fdec(maxenc[(size_t)dn * H + h]);
    const float p = expf(logits[t] - m);
    logits[t] = p;
    atomicAdd(&ssum[(size_t)dn * H + h], p);
}

// ---------------------------------------------------------------------------
// Edge pass 3: out[dst,h,:] += (p / (s[dst,h]+1e-9)) * hs[src,h,:]
// ---------------------------------------------------------------------------
template <int H, int D>
__global__ void edge_scatter_kernel(const int* __restrict__ src,
                                    const int* __restrict__ dst,
                                    const float* __restrict__ hs,
                                    const float* __restrict__ p,
                                    const float* __restrict__ ssum,
                                    float* __restrict__ out,
                                    int E) {
    const long long t = (long long)blockIdx.x * blockDim.x + threadIdx.x;
    if (t >= (long long)E * H) return;
    const int e = (int)(t / H);
    const int h = (int)(t % H);
    const int sn = src[e];
    const int dn = dst[e];
    const float alpha = p[t] / (ssum[(size_t)dn * H + h] + 1e-9f);
    const float* __restrict__ ps = hs + ((size_t)sn * H + h) * D;
    float* __restrict__ po = out + ((size_t)dn * H + h) * D;
    #pragma unroll
    for (int i = 0; i < D; ++i) atomicAdd(&po[i], alpha * ps[i]);
}

// ---------------------------------------------------------------------------
// Elementwise helpers
// ---------------------------------------------------------------------------
__global__ void elu_kernel(float* __restrict__ x, long long n) {
    const long long t = (long long)blockIdx.x * blockDim.x + threadIdx.x;
    if (t >= n) return;
    const float v = x[t];
    x[t] = (v > 0.0f) ? v : (expf(v) - 1.0f);
}

__global__ void fill_f32_kernel(float* __restrict__ p, float v, long long n) {
    const long long t = (long long)blockIdx.x * blockDim.x + threadIdx.x;
    if (t < n) p[t] = v;
}

__global__ void fill_u32_kernel(unsigned* __restrict__ p, unsigned v, long long n) {
    const long long t = (long long)blockIdx.x * blockDim.x + threadIdx.x;
    if (t < n) p[t] = v;
}

// ---------------------------------------------------------------------------
// Host-side launch
// ---------------------------------------------------------------------------
static inline int blocks_for(long long n, int bs) { return (int)((n + bs - 1) / bs); }

extern "C" void kernel_launch(void* const* d_in, const int* in_sizes, int n_in,
                              void* d_out, int out_size, void* d_ws, size_t ws_size,
                              hipStream_t stream) {
    const float* x   = (const float*)d_in[0];
    const int*   src = (const int*)  d_in[1];
    const int*   dst = (const int*)  d_in[2];
    const float* W1s = (const float*)d_in[3];
    const float* b1s = (const float*)d_in[4];
    const float* W1d = (const float*)d_in[5];
    const float* b1d = (const float*)d_in[6];
    const float* a1  = (const float*)d_in[7];
    const float* W2s = (const float*)d_in[8];
    const float* b2s = (const float*)d_in[9];
    const float* W2d = (const float*)d_in[10];
    const float* b2d = (const float*)d_in[11];
    const float* a2  = (const float*)d_in[12];

    const int F_IN = 512, H1 = 8, HD = 64, C = 40;
    const int Nn = in_sizes[0] / F_IN;     // 100000
    const int E  = in_sizes[1];            // 1600000
    float* out2 = (float*)d_out;           // [Nn, 40]

    // ---- workspace layout (floats) ----
    float* f = (float*)d_ws;
    float*    hs1     = f;                 f += (size_t)Nn * HD;
    float*    hd1     = f;                 f += (size_t)Nn * HD;
    float*    out1    = f;                 f += (size_t)Nn * HD;
    float*    logits1 = f;                 f += (size_t)E * H1;
    unsigned* maxenc1 = (unsigned*)f;      f += (size_t)Nn * H1;
    float*    s1      = f;                 f += (size_t)Nn * H1;
    // layer-2 scratch aliases the hs1/hd1 region (dead after layer-1 scatter)
    float*    hs2     = hs1;                                  // Nn*C
    float*    hd2     = hs1 + (size_t)Nn * C;                 // Nn*C
    float*    logits2 = hs1 + (size_t)Nn * 2 * C;             // E
    unsigned* maxenc2 = (unsigned*)(logits2 + E);             // Nn
    float*    s2      = (float*)(maxenc2) + Nn;               // Nn

    const dim3 gemm_blk(GEMM_WAVES * 32);
    const int  gemm_gx = blocks_for(Nn / 16, GEMM_WAVES);

    // ================= Layer 1 (NCOLS=64, K=512; RS=68, 2 bufs) =============
    {
        const size_t shmem = (size_t)2 * 128 * 68 * sizeof(float);   // 68 KB
        wmma_gemm_bias<64, 512><<<gemm_gx, gemm_blk, shmem, stream>>>(x, W1s, b1s, hs1, Nn);
        wmma_gemm_bias<64, 512><<<gemm_gx, gemm_blk, shmem, stream>>>(x, W1d, b1d, hd1, Nn);
    }
    fill_u32_kernel<<<blocks_for((long long)Nn * H1, 256), 256, 0, stream>>>(maxenc1, ENC_NEG_INF, (long long)Nn * H1);
    fill_f32_kernel<<<blocks_for((long long)Nn * H1, 256), 256, 0, stream>>>(s1, 0.0f, (long long)Nn * H1);
    fill_f32_kernel<<<blocks_for((long long)Nn * HD, 256), 256, 0, stream>>>(out1, 0.0f, (long long)Nn * HD);

    {
        const long long tot = (long long)E * H1;
        edge_logits_kernel<8, 8><<<blocks_for(tot, 256), 256, 0, stream>>>(src, dst, hs1, hd1, a1, logits1, maxenc1, E);
        edge_expsum_kernel<8><<<blocks_for(tot, 256), 256, 0, stream>>>(dst, maxenc1, logits1, s1, E);
        edge_scatter_kernel<8, 8><<<blocks_for(tot, 256), 256, 0, stream>>>(src, dst, hs1, logits1, s1, out1, E);
    }
    elu_kernel<<<blocks_for((long long)Nn * HD, 256), 256, 0, stream>>>(out1, (long long)Nn * HD);

    // ================= Layer 2 (NCOLS=40, K=64; RS=40, 1 chunk) =============
    {
        const size_t shmem = (size_t)2 * 64 * 40 * sizeof(float);    // 20 KB
        wmma_gemm_bias<40, 64><<<gemm_gx, gemm_blk, shmem, stream>>>(out1, W2s, b2s, hs2, Nn);
        wmma_gemm_bias<40, 64><<<gemm_gx, gemm_blk, shmem, stream>>>(out1, W2d, b2d, hd2, Nn);
    }
    fill_u32_kernel<<<blocks_for((long long)Nn, 256), 256, 0, stream>>>(maxenc2, ENC_NEG_INF, (long long)Nn);
    fill_f32_kernel<<<blocks_for((long long)Nn, 256), 256, 0, stream>>>(s2, 0.0f, (long long)Nn);
    fill_f32_kernel<<<blocks_for((long long)Nn * C, 256), 256, 0, stream>>>(out2, 0.0f, (long long)Nn * C);

    {
        const long long tot = (long long)E;     // H = 1
        edge_logits_kernel<1, 40><<<blocks_for(tot, 256), 256, 0, stream>>>(src, dst, hs2, hd2, a2, logits2, maxenc2, E);
        edge_expsum_kernel<1><<<blocks_for(tot, 256), 256, 0, stream>>>(dst, maxenc2, logits2, s2, E);
        edge_scatter_kernel<1, 40><<<blocks_for(tot, 256), 256, 0, stream>>>(src, dst, hs2, logits2, s2, out2, E);
    }
}